// IOU_89928025244080
// MI455X (gfx1250) — compile-verified
//
#include <hip/hip_runtime.h>
#include <stdint.h>

#define NBLK 1024
#define TPB  256
#define NT   100   // number of thresholds
#define NBIN 101   // bins 0..100

typedef __attribute__((ext_vector_type(2))) float v2f;
typedef __attribute__((ext_vector_type(8))) float v8f;

__device__ __forceinline__ unsigned lds_off(const void* p) {
  // Generic pointers to LDS carry the allocation-relative offset in the low 32 bits.
  return (unsigned)(uintptr_t)p;
}

// bin = #{k in [0,100) : (float)(k*0.01f) < p}  == searchsorted(thr, p, 'left')
__device__ __forceinline__ int bin_of(float p) {
  int c = (int)floorf(p * 100.0f);
  c = c < 0 ? 0 : (c > 100 ? 100 : c);
  while (c < 100 && ((float)c * 0.01f) < p) ++c;       // at most ~1 iter
  while (c > 0 && ((float)(c - 1) * 0.01f) >= p) --c;  // at most ~1 iter
  return c;
}

__global__ void zero_ws(unsigned* ws) {
  int t = blockIdx.x * blockDim.x + threadIdx.x;
  if (t < 2 * NBIN) ws[t] = 0u;
}

__global__ __launch_bounds__(TPB)
void hist_kernel(const float* __restrict__ pred, const int* __restrict__ gt,
                 unsigned* __restrict__ gpos, unsigned* __restrict__ gall,
                 int iters, long long tail_start, long long N) {
  __shared__ unsigned hist[NBIN];
  __shared__ float4 bufP[2][TPB];
  __shared__ int4   bufG[2][TPB];

  const int t = threadIdx.x;
  const int b = blockIdx.x;
  for (int i = t; i < NBIN; i += TPB) hist[i] = 0u;
  __syncthreads();

  const long long blockBase = (long long)b * ((long long)iters * TPB * 4);

  auto issue = [&](int it) {
    int s = it & 1;
    long long base = blockBase + (long long)it * (TPB * 4) + (long long)t * 4;
    unsigned lp = lds_off(&bufP[s][t]);
    unsigned lg = lds_off(&bufG[s][t]);
    unsigned long long gp = (unsigned long long)(const void*)(pred + base);
    unsigned long long gg = (unsigned long long)(const void*)(gt + base);
    asm volatile("global_load_async_to_lds_b128 %0, %1, off" :: "v"(lp), "v"(gp) : "memory");
    asm volatile("global_load_async_to_lds_b128 %0, %1, off" :: "v"(lg), "v"(gg) : "memory");
  };

  if (iters > 0) {
    issue(0);
    for (int it = 0; it < iters; ++it) {
      if (it + 1 < iters) {
        issue(it + 1);
        asm volatile("s_wait_asynccnt 0x2" ::: "memory");  // stage `it` landed
      } else {
        asm volatile("s_wait_asynccnt 0x0" ::: "memory");
      }
      int s = it & 1;
      float4 p4 = bufP[s][t];
      int4   g4 = bufG[s][t];
      atomicAdd(&hist[bin_of(p4.x)], 0x10000u | (unsigned)(g4.x & 1));
      atomicAdd(&hist[bin_of(p4.y)], 0x10000u | (unsigned)(g4.y & 1));
      atomicAdd(&hist[bin_of(p4.z)], 0x10000u | (unsigned)(g4.z & 1));
      atomicAdd(&hist[bin_of(p4.w)], 0x10000u | (unsigned)(g4.w & 1));
    }
  }
  // scalar tail (empty for N = 2^25)
  for (long long i = tail_start + (long long)(b * TPB + t); i < N;
       i += (long long)NBLK * TPB) {
    atomicAdd(&hist[bin_of(pred[i])], 0x10000u | (unsigned)(gt[i] & 1));
  }
  __syncthreads();
  if (t < NBIN) {
    unsigned v = hist[t];
    atomicAdd(&gpos[t], v & 0xFFFFu);
    atomicAdd(&gall[t], v >> 16);
  }
}

// Suffix sums t[k] = sum_{c>k} h[c] as strictly-upper-triangular matvec on WMMA.
__global__ __launch_bounds__(32)
void finalize_kernel(const unsigned* __restrict__ gpos,
                     const unsigned* __restrict__ gall, float* __restrict__ out) {
  __shared__ float hp[104], ha[104];
  const int lane = threadIdx.x;
  for (int i = lane; i < 104; i += 32) {
    hp[i] = (i < NBIN) ? (float)gpos[i] : 0.0f;
    ha[i] = (i < NBIN) ? (float)gall[i] : 0.0f;
  }
  __syncthreads();

  unsigned ngt_u = 0;
  for (int c = 0; c < NBIN; ++c) ngt_u += gpos[c];
  const float ngt = (float)ngt_u;

#if __has_builtin(__builtin_amdgcn_wmma_f32_16x16x4_f32)
  const int halfSel = (lane < 16) ? 0 : 2;  // A/B layout: lanes 16-31 carry K+2
  const int n15 = lane & 15;
  for (int j = 0; j < 7; ++j) {
    const int base = 16 * j;
    v8f accP = {0.f, 0.f, 0.f, 0.f, 0.f, 0.f, 0.f, 0.f};
    v8f accA = {0.f, 0.f, 0.f, 0.f, 0.f, 0.f, 0.f, 0.f};
    for (int g = 0; g < 26; ++g) {  // K = 0..103, h[101..103]=0
      const int k0 = 4 * g + halfSel;
      v2f ap, aa, bb;
      ap[0] = hp[k0];     ap[1] = hp[k0 + 1];   // A[m][k] = h[4g+k], all m
      aa[0] = ha[k0];     aa[1] = ha[k0 + 1];
      const int n = base + n15;                 // B[k][n] = (c > thresh_idx)
      bb[0] = (k0 > n) ? 1.0f : 0.0f;
      bb[1] = (k0 + 1 > n) ? 1.0f : 0.0f;
      accP = __builtin_amdgcn_wmma_f32_16x16x4_f32(false, ap, false, bb,
                                                   (short)0, accP, false, false);
      accA = __builtin_amdgcn_wmma_f32_16x16x4_f32(false, aa, false, bb,
                                                   (short)0, accA, false, false);
    }
    const int k = base + n15;
    if (lane < 16 && k < NT) {
      float tp = accP[0];  // D row M=0 (result replicated over rows)
      float pp = accA[0];
      float un = pp + ngt - tp;
      out[k] = (float)k * 0.01f;
      out[NT + k] = (un > 0.0f) ? (tp / fmaxf(un, 1.0f)) : 0.0f;
    }
  }
#else
  for (int k = lane; k < NT; k += 32) {
    float tp = 0.f, pp = 0.f;
    for (int c = k + 1; c < NBIN; ++c) { tp += hp[c]; pp += ha[c]; }
    float un = pp + ngt - tp;
    out[k] = (float)k * 0.01f;
    out[NT + k] = (un > 0.0f) ? (tp / fmaxf(un, 1.0f)) : 0.0f;
  }
#endif
}

extern "C" void kernel_launch(void* const* d_in, const int* in_sizes, int n_in,
                              void* d_out, int out_size, void* d_ws, size_t ws_size,
                              hipStream_t stream) {
  (void)n_in; (void)out_size; (void)ws_size;
  const float* pred = (const float*)d_in[0];
  const int*   gt   = (const int*)d_in[1];
  const long long N = (long long)in_sizes[0];

  unsigned* gpos = (unsigned*)d_ws;
  unsigned* gall = gpos + NBIN;
  float* out = (float*)d_out;

  zero_ws<<<1, 256, 0, stream>>>(gpos);

  const long long chunk = (long long)NBLK * TPB * 4;  // pixels per full sweep
  int iters = (int)(N / chunk);                       // 32 for N = 2^25
  long long tail_start = (long long)iters * chunk;

  hist_kernel<<<NBLK, TPB, 0, stream>>>(pred, gt, gpos, gall, iters, tail_start, N);
  finalize_kernel<<<1, 32, 0, stream>>>(gpos, gall, out);
}